// _ECELoss_3891240370496
// MI455X (gfx1250) — compile-verified
//
#include <hip/hip_runtime.h>
#include <math.h>

#define NBINS 15
#define CPS   10          // classes per sample
#define TPB   256         // threads per block
#define TILE  256         // samples per tile (== TPB, one sample per thread)
#define TILE_DW (TILE * CPS)        // dwords per array per tile (2560)
#define TILE_CHUNKS (TILE_DW / 4)   // b128 chunks per array per tile (640)

// ---- CDNA5 async global->LDS path (gfx1250), guarded so the host pass and
// ---- older toolchains still compile via a synchronous b128 staging fallback.
#if defined(__has_builtin)
#  if __has_builtin(__builtin_amdgcn_global_load_async_to_lds_b128)
#    define HAVE_ASYNC_B128 1
#  endif
#endif

typedef __attribute__((ext_vector_type(4))) int v4i;
#if defined(HAVE_ASYNC_B128)
typedef __attribute__((address_space(1))) v4i* gptr128_t;   // global (printed "__device__")
typedef __attribute__((address_space(3))) v4i* lptr128_t;   // LDS
#  define CPOL_NT 1   // gfx12+ cpol: TH[2:0]=1 -> non-temporal (read-once 320MB stream > 192MB L2)
#endif

static __device__ __forceinline__ void wait_async_copies() {
#if defined(HAVE_ASYNC_B128)
#  if __has_builtin(__builtin_amdgcn_s_wait_asynccnt)
    __builtin_amdgcn_s_wait_asynccnt(0);
#  else
    asm volatile("s_wait_asynccnt 0" ::: "memory");
#  endif
#endif
}

static __device__ __forceinline__ void async_copy_b128(const void* gsrc, void* ldst) {
#if defined(HAVE_ASYNC_B128)
  v4i* gp = (v4i*)gsrc;   // strip const, reinterpret
  v4i* lp = (v4i*)ldst;
  __builtin_amdgcn_global_load_async_to_lds_b128((gptr128_t)gp, (lptr128_t)lp, 0, CPOL_NT);
#else
  *(v4i*)ldst = *(const v4i*)gsrc;
#endif
}

// Per-sample math: conf = sigmoid(max logit); pred = first c with logit>=0 (else 0);
// label_idx = first-occurrence argmax of labels; bin by (lower, upper] membership.
static __device__ __forceinline__ void process_sample(
    const float* __restrict__ lrow, const int* __restrict__ labrow,
    float* s_conf, float* s_acc, unsigned* s_cnt) {
  float m = -INFINITY;
  int pred = -1;
#pragma unroll
  for (int c = 0; c < CPS; ++c) {
    float v = lrow[c];
    m = fmaxf(m, v);
    if (pred < 0 && v >= 0.0f) pred = c;
  }
  if (pred < 0) pred = 0;

  int li = 0;
  int bm = labrow[0];
#pragma unroll
  for (int c = 1; c < CPS; ++c) {
    int v = labrow[c];
    if (v > bm) { bm = v; li = c; }
  }

  float conf = 1.0f / (1.0f + expf(-m));
  if (conf > 0.0f) {                       // bin membership is (lower, upper]
    int bin = (int)ceilf(conf * (float)NBINS) - 1;
    bin = bin < 0 ? 0 : (bin > NBINS - 1 ? NBINS - 1 : bin);
    atomicAdd(&s_cnt[bin], 1u);
    atomicAdd(&s_conf[bin], conf);
    atomicAdd(&s_acc[bin], (pred == li) ? 1.0f : 0.0f);
  }
}

__global__ void __launch_bounds__(TPB)
ece_main(const float* __restrict__ logits, const int* __restrict__ labels,
         unsigned* __restrict__ cnt_ws, float* __restrict__ conf_ws,
         float* __restrict__ acc_ws, long long nTiles, long long N) {
  __shared__ float s_logit[2][TILE_DW];
  __shared__ int   s_label[2][TILE_DW];
  __shared__ float s_conf[NBINS];
  __shared__ float s_acc[NBINS];
  __shared__ unsigned s_cnt[NBINS];

  const int tid = threadIdx.x;
  if (tid < NBINS) { s_conf[tid] = 0.0f; s_acc[tid] = 0.0f; s_cnt[tid] = 0u; }
  __syncthreads();

  auto issue_copy = [&](int buf, long long t) {
    const float* gL = logits + t * (long long)TILE_DW;
    const int*   gI = labels + t * (long long)TILE_DW;
    for (int idx = tid; idx < TILE_CHUNKS; idx += TPB) {
      async_copy_b128(gL + idx * 4, &s_logit[buf][idx * 4]);
      async_copy_b128(gI + idx * 4, &s_label[buf][idx * 4]);
    }
  };

  const long long t0     = blockIdx.x;
  const long long stride = gridDim.x;
  int cur = 0;

  if (t0 < nTiles) {
    issue_copy(0, t0);
    wait_async_copies();
    __syncthreads();
  }

  for (long long t = t0; t < nTiles; t += stride) {
    long long nt = t + stride;
    if (nt < nTiles) issue_copy(cur ^ 1, nt);   // overlap copy(t+stride) with compute(t)

    process_sample(&s_logit[cur][tid * CPS], &s_label[cur][tid * CPS],
                   s_conf, s_acc, s_cnt);

    wait_async_copies();   // next buffer complete in this wave
    __syncthreads();       // all waves done reading cur + all copies visible
    cur ^= 1;
  }

  // Tail samples (N not multiple of TILE): straight from global.
  for (long long s = nTiles * TILE + (long long)blockIdx.x * TPB + tid;
       s < N; s += stride * TPB) {
    process_sample(logits + s * CPS, labels + s * CPS, s_conf, s_acc, s_cnt);
  }

  __syncthreads();
  if (tid < NBINS) {
    if (s_cnt[tid])  atomicAdd(&cnt_ws[tid],  s_cnt[tid]);
    if (s_conf[tid] != 0.0f) atomicAdd(&conf_ws[tid], s_conf[tid]);
    if (s_acc[tid]  != 0.0f) atomicAdd(&acc_ws[tid],  s_acc[tid]);
  }
}

__global__ void ece_zero_ws(unsigned* ws) {
  if (threadIdx.x < 3 * NBINS + 3) ws[threadIdx.x] = 0u;
}

__global__ void ece_final(const unsigned* __restrict__ cnt,
                          const float* __restrict__ conf,
                          const float* __restrict__ acc,
                          float* __restrict__ out, float invN) {
  if (threadIdx.x == 0 && blockIdx.x == 0) {
    float e = 0.0f;
    for (int b = 0; b < NBINS; ++b) {
      unsigned c = cnt[b];
      if (c > 0u) {
        float safe = fmaxf((float)c, 1.0f);
        float gap  = fabsf(conf[b] / safe - acc[b] / safe);
        e += gap * ((float)c * invN);
      }
    }
    out[0] = e;
  }
}

extern "C" void kernel_launch(void* const* d_in, const int* in_sizes, int n_in,
                              void* d_out, int out_size, void* d_ws, size_t ws_size,
                              hipStream_t stream) {
  const float* logits = (const float*)d_in[0];
  const int*   labels = (const int*)d_in[1];
  const long long total = (long long)in_sizes[0];   // N * C
  const long long N     = total / CPS;

  // ws layout (dwords): [0,15) counts, [16,31) conf sums, [32,47) acc sums
  unsigned* cnt_ws  = (unsigned*)d_ws;
  float*    conf_ws = (float*)d_ws + 16;
  float*    acc_ws  = (float*)d_ws + 32;

  const long long nTiles = N / TILE;
  long long wantBlocks = nTiles > 0 ? nTiles : 1;
  int grid = (int)(wantBlocks < 4096 ? wantBlocks : 4096);

  ece_zero_ws<<<1, 64, 0, stream>>>((unsigned*)d_ws);
  ece_main<<<grid, TPB, 0, stream>>>(logits, labels, cnt_ws, conf_ws, acc_ws,
                                     nTiles, N);
  ece_final<<<1, 32, 0, stream>>>(cnt_ws, conf_ws, acc_ws, (float*)d_out,
                                  1.0f / (float)N);
}